// ToHeteroLinear_21028159881437
// MI455X (gfx1250) — compile-verified
//
#include <hip/hip_runtime.h>

// f32 WMMA operand/accumulator vector types (CDNA5, wave32)
typedef float v2f __attribute__((ext_vector_type(2)));
typedef float v4f __attribute__((ext_vector_type(4)));
typedef float v8f __attribute__((ext_vector_type(8)));

#define NROWS   131072
#define KDIM    256
#define ODIM    256
#define NTYPES  8

#define ROWS_PER_BLOCK 128         // 8 row-subtiles of 16 per wave
#define NSUB           8
#define LDS_STRIDE     260         // 256 + 4 floats pad -> conflict-free A reads
#define LDS_STRIDE4    65          // in float4 units

// Block: 512 threads = 16 waves. Each block computes a 128-row x 256-col output
// stripe; wave w computes columns [16w,16w+16) for all 8 row subtiles, sharing
// each B fragment across 8 WMMAs (register blocking on M).
__global__ __launch_bounds__(512) void hetero_linear_wmma_f32(
    const float* __restrict__ x,        // [N, 256]
    const int*   __restrict__ type_vec, // [N], sorted, values 0..7
    const float* __restrict__ weight,   // [8, 256, 256]
    const float* __restrict__ bias,     // [8, 256]
    float*       __restrict__ out)      // [N, 256]
{
    __shared__ float sA[ROWS_PER_BLOCK * LDS_STRIDE];  // ~130 KB padded A tile
    __shared__ int   sT[ROWS_PER_BLOCK];

    const int tid  = threadIdx.x;
    const int lane = tid & 31;
    const int wave = tid >> 5;                 // 0..15 -> column tile
    const int row0 = blockIdx.x * ROWS_PER_BLOCK;

    // ---- Stage A tile: 128 rows x 256 f32 = 8192 float4, 16 per thread ----
    {
        const v4f* src = (const v4f*)(x + (size_t)row0 * KDIM);
        v4f*       dst = (v4f*)sA;
#pragma unroll
        for (int i = 0; i < 16; ++i) {
            const int flat = tid + i * 512;    // 0..8191
            const int r    = flat >> 6;        // row (64 float4 per row)
            const int c4   = flat & 63;        // float4 col
            // x is streamed exactly once -> non-temporal
            dst[r * LDS_STRIDE4 + c4] = __builtin_nontemporal_load(&src[flat]);
        }
    }
    if (tid < ROWS_PER_BLOCK) sT[tid] = type_vec[row0 + tid];
    __syncthreads();

    // f32 WMMA 16x16x4 lane mapping:
    //   A: lane L holds A[m][kb], A[m][kb+1], m = L&15, kb = k0 + 2*(L>=16)
    //   B: lane L holds B[kb][n], B[kb+1][n],  n = n0 + (L&15)
    //   C/D: VGPR r <-> row m = r + 8*(L>=16), col n = n0 + (L&15)
    const int m     = lane & 15;
    const int khalf = (lane >> 4) << 1;        // 0 or 2
    const int mhalf = (lane >> 4) << 3;        // 0 or 8
    const int n     = (wave << 4) + m;         // global output column

    // type_vec sorted => row types within the block bounded by the ends
    const int tmin = sT[0];
    const int tmax = sT[ROWS_PER_BLOCK - 1];
    const bool uniformType = (tmin == tmax);

    // Per-lane packed row types (nibble per row) for the rare mixed-type path:
    // 8 u32 regs instead of 64 live compare operands.
    unsigned int packedT[NSUB];
#pragma unroll
    for (int s = 0; s < NSUB; ++s) {
        unsigned int p = 0;
#pragma unroll
        for (int r = 0; r < 8; ++r)
            p |= ((unsigned int)sT[s * 16 + mhalf + r] & 0xF) << (4 * r);
        packedT[s] = p;
    }

    // Uniform SGPR base + small per-lane element offset => single-VGPR stores
    // with immediate offsets (no 64-bit per-element addresses to spill).
    float*    outB = out + (size_t)row0 * ODIM;
    const int boff = mhalf * ODIM + n;

    for (int t = tmin; t <= tmax; ++t) {
        const float* W  = weight + (size_t)t * (KDIM * ODIM);
        const float  bv = bias[t * ODIM + n];

        v8f c[NSUB];
#pragma unroll
        for (int s = 0; s < NSUB; ++s) {
            // Fold bias into the accumulator init (just reorders the f32 sum)
            c[s] = (v8f){bv, bv, bv, bv, bv, bv, bv, bv};
        }

#pragma unroll 1
        for (int k0 = 0; k0 < KDIM; k0 += 4) {
            const int kb = k0 + khalf;
            // One B fragment feeds 8 WMMAs (8 row subtiles)
            v2f b;
            b.x = W[kb * ODIM + n];
            b.y = W[(kb + 1) * ODIM + n];
            // Batch the 8 A-fragment LDS loads, then the 8 WMMAs, so the
            // compiler stages s_wait_dscnt 7..0 instead of serializing.
            v2f a[NSUB];
#pragma unroll
            for (int s = 0; s < NSUB; ++s) {
                a[s] = *(const v2f*)&sA[(m + 16 * s) * LDS_STRIDE + kb];
            }
#pragma unroll
            for (int s = 0; s < NSUB; ++s) {
                c[s] = __builtin_amdgcn_wmma_f32_16x16x4_f32(
                           false, a[s], false, b, (short)0, c[s], false, false);
            }
        }

        if (uniformType) {
            // Fast path (almost all blocks): one type covers all 128 rows ->
            // unconditional NT stores, shared address VGPR + immediate offsets.
#pragma unroll
            for (int s = 0; s < NSUB; ++s)
#pragma unroll
                for (int r = 0; r < 8; ++r)
                    __builtin_nontemporal_store(
                        c[s][r], &outB[boff + (s * 16 + r) * ODIM]);
        } else {
            // Rare mixed-type block: per-row predicate from packed nibbles.
            // Each row's type is in [tmin,tmax] so it is stored exactly once.
#pragma unroll
            for (int s = 0; s < NSUB; ++s)
#pragma unroll
                for (int r = 0; r < 8; ++r)
                    if ((int)((packedT[s] >> (4 * r)) & 0xF) == t)
                        __builtin_nontemporal_store(
                            c[s][r], &outB[boff + (s * 16 + r) * ODIM]);
        }
    }
}

extern "C" void kernel_launch(void* const* d_in, const int* in_sizes, int n_in,
                              void* d_out, int out_size, void* d_ws, size_t ws_size,
                              hipStream_t stream) {
    const float* x        = (const float*)d_in[0];
    const int*   type_vec = (const int*)  d_in[1];
    const float* weight   = (const float*)d_in[2];
    const float* bias     = (const float*)d_in[3];
    float*       out      = (float*)d_out;

    dim3 grid(NROWS / ROWS_PER_BLOCK);   // 1024 blocks, 128 rows each
    dim3 block(512);                     // 16 waves of 32
    hetero_linear_wmma_f32<<<grid, block, 0, stream>>>(x, type_vec, weight, bias, out);
}